// UserGNNEncoder_61718680044159
// MI455X (gfx1250) — compile-verified
//
#include <hip/hip_runtime.h>

typedef float v2f __attribute__((ext_vector_type(2)));
typedef float v8f __attribute__((ext_vector_type(8)));

// ---------------------------------------------------------------------------
// Utility kernels
// ---------------------------------------------------------------------------
__global__ void zero_f32(float* __restrict__ p, int n4) {
  int i = blockIdx.x * blockDim.x + threadIdx.x;
  if (i < n4) ((float4*)p)[i] = make_float4(0.f, 0.f, 0.f, 0.f);
}

__global__ void degree_count(const long long* __restrict__ dst,
                             float* __restrict__ deg, int E) {
  int i = blockIdx.x * blockDim.x + threadIdx.x;
  if (i < E) atomicAdd(&deg[(int)dst[i]], 1.0f);
}

// Scatter-add of x[src] into agg[dst]; one thread handles 4 contiguous channels.
// dim4 = dim/4 = 1<<cshift.
__global__ void scatter_add4(const float* __restrict__ x,
                             const long long* __restrict__ esrc,
                             const long long* __restrict__ edst,
                             float* __restrict__ agg,
                             int E, int cshift, int dim) {
  int gid = blockIdx.x * blockDim.x + threadIdx.x;
  int dim4 = 1 << cshift;
  int total = E << cshift;
  if (gid >= total) return;
  int e = gid >> cshift;
  int c = (gid & (dim4 - 1)) << 2;
  int s = (int)esrc[e];
  int d = (int)edst[e];
  const float4 v = *(const float4*)(x + (size_t)s * dim + c);
  float* p = agg + (size_t)d * dim + c;
  atomicAdd(p + 0, v.x);
  atomicAdd(p + 1, v.y);
  atomicAdd(p + 2, v.z);
  atomicAdd(p + 3, v.w);
}

// agg[row, :] *= 1 / max(deg[row], 1)
__global__ void row_scale(float* __restrict__ agg, const float* __restrict__ deg,
                          int rows, int cshift, int dim) {
  int gid = blockIdx.x * blockDim.x + threadIdx.x;
  int dim4 = 1 << cshift;
  int total = rows << cshift;
  if (gid >= total) return;
  int row = gid >> cshift;
  int c = (gid & (dim4 - 1)) << 2;
  float r = 1.0f / fmaxf(deg[row], 1.0f);
  float4* p = (float4*)(agg + (size_t)row * dim + c);
  float4 v = *p;
  v.x *= r; v.y *= r; v.z *= r; v.w *= r;
  *p = v;
}

// ---------------------------------------------------------------------------
// Fused dual-GEMM:  out = act( A @ Wl^T + X @ Wr^T + bias )
//   A,X: [M,K] row-major     Wl,Wr: [N,K] row-major     out: [M,N]
// Each wave computes a 16x64 strip via 4x V_WMMA_F32_16X16X4_F32 tiles.
// Requires: M%16==0, N%64==0, K%4==0.
// ---------------------------------------------------------------------------
__device__ __forceinline__ void wmma_pair(const float* __restrict__ P,
                                          const float* __restrict__ W,
                                          int K, int row, int colBase,
                                          int laneHalf, int laneLo,
                                          v8f& c0, v8f& c1, v8f& c2, v8f& c3) {
  // A 16x4 f32 layout: lane<16 -> (K0,K1), lane>=16 -> (K2,K3) of row M=lane%16
  // B 4x16 f32 layout: lane<16 -> (K0,K1), lane>=16 -> (K2,K3) of col N=lane%16
  const float* pa  = P + (size_t)row * K + (laneHalf << 1);
  const float* pb0 = W + (size_t)(colBase + 0  + laneLo) * K + (laneHalf << 1);
  const float* pb1 = W + (size_t)(colBase + 16 + laneLo) * K + (laneHalf << 1);
  const float* pb2 = W + (size_t)(colBase + 32 + laneLo) * K + (laneHalf << 1);
  const float* pb3 = W + (size_t)(colBase + 48 + laneLo) * K + (laneHalf << 1);
#pragma unroll 4
  for (int k = 0; k < K; k += 4) {
    v2f a  = *(const v2f*)(pa  + k);
    v2f b0 = *(const v2f*)(pb0 + k);
    v2f b1 = *(const v2f*)(pb1 + k);
    v2f b2 = *(const v2f*)(pb2 + k);
    v2f b3 = *(const v2f*)(pb3 + k);
    c0 = __builtin_amdgcn_wmma_f32_16x16x4_f32(false, a, false, b0, (short)0, c0, false, false);
    c1 = __builtin_amdgcn_wmma_f32_16x16x4_f32(false, a, false, b1, (short)0, c1, false, false);
    c2 = __builtin_amdgcn_wmma_f32_16x16x4_f32(false, a, false, b2, (short)0, c2, false, false);
    c3 = __builtin_amdgcn_wmma_f32_16x16x4_f32(false, a, false, b3, (short)0, c3, false, false);
  }
}

__global__ __launch_bounds__(256)
void sage_gemm(const float* __restrict__ A,   // [M,K] aggregated means
               const float* __restrict__ Wl,  // [N,K]
               const float* __restrict__ X,   // [M,K] dst features (may be null)
               const float* __restrict__ Wr,  // [N,K] (may be null)
               const float* __restrict__ bias,// [N]
               float* __restrict__ out,       // [M,N]
               int M, int N, int K, int doRelu) {
  int wavesPerBlock = blockDim.x >> 5;
  int w = blockIdx.x * wavesPerBlock + (int)(threadIdx.x >> 5);
  int nStrips = N >> 6;                    // strips of 64 cols
  int totalStrips = (M >> 4) * nStrips;
  if (w >= totalStrips) return;            // wave-uniform: EXEC stays all-ones
  int mtile = w / nStrips;
  int nstrip = w - mtile * nStrips;

  int lane = threadIdx.x & 31;
  int laneHalf = lane >> 4;
  int laneLo = lane & 15;

  v8f c0 = {}, c1 = {}, c2 = {}, c3 = {};

  int row = (mtile << 4) + laneLo;
  int colBase = nstrip << 6;

  wmma_pair(A, Wl, K, row, colBase, laneHalf, laneLo, c0, c1, c2, c3);
  if (X) wmma_pair(X, Wr, K, row, colBase, laneHalf, laneLo, c0, c1, c2, c3);

  // C/D 16x16 f32 layout: element (vgpr r, lane) = C[r + 8*laneHalf, laneLo]
  int rowBase = (mtile << 4) + (laneHalf << 3);
  v8f acc[4] = {c0, c1, c2, c3};
#pragma unroll
  for (int t = 0; t < 4; ++t) {
    int col = colBase + (t << 4) + laneLo;
    float bv = bias[col];
    v8f cc = acc[t];
#pragma unroll
    for (int r = 0; r < 8; ++r) {
      float v = cc[r] + bv;
      if (doRelu) v = fmaxf(v, 0.0f);
      out[(size_t)(rowBase + r) * N + col] = v;
    }
  }
}

// ---------------------------------------------------------------------------
// Host-side orchestration
// ---------------------------------------------------------------------------
static inline int cdiv(int a, int b) { return (a + b - 1) / b; }

extern "C" void kernel_launch(void* const* d_in, const int* in_sizes, int n_in,
                              void* d_out, int out_size, void* d_ws, size_t ws_size,
                              hipStream_t stream) {
  const float*     x_product  = (const float*)d_in[0];
  const float*     x_customer = (const float*)d_in[1];
  const long long* edge_pp    = (const long long*)d_in[2];
  const long long* edge_pc    = (const long long*)d_in[3];
  const float* Wl1 = (const float*)d_in[4];
  const float* bl1 = (const float*)d_in[5];
  const float* Wr1 = (const float*)d_in[6];
  const float* Wl2 = (const float*)d_in[7];
  const float* bl2 = (const float*)d_in[8];
  const float* Wr2 = (const float*)d_in[9];
  const float* Wl3 = (const float*)d_in[10];
  const float* bl3 = (const float*)d_in[11];
  const float* Wr3 = (const float*)d_in[12];
  const float* Wlin = (const float*)d_in[13];
  const float* blin = (const float*)d_in[14];

  const int NP = 20000, NC = 20000, D = 256, H = 512, O = 256;
  const int E = in_sizes[2] / 2;

  // workspace layout (floats):
  //  [mean1 NP*D][mean2 NP*D]   <- this 2*NP*D region is later reused as mean3 (NC*H)
  //  [prod  NP*H]               <- later reused as cust2 (NC*H)
  //  [cust  NC*H]
  //  [deg_pp NP][deg_pc NC]
  float* ws     = (float*)d_ws;
  float* mean1  = ws;
  float* mean2  = ws + (size_t)NP * D;
  float* prod   = ws + (size_t)2 * NP * D;
  float* cust   = prod + (size_t)NP * H;
  float* deg_pp = cust + (size_t)NC * H;
  float* deg_pc = deg_pp + NP;

  const long long* pp_src = edge_pp;      const long long* pp_dst = edge_pp + E;
  const long long* pc_src = edge_pc;      const long long* pc_dst = edge_pc + E;

  const int T = 256;

  // ---- degrees + mean aggregation (D=256, cshift=6) ----
  zero_f32<<<cdiv(2 * NP * D / 4, T), T, 0, stream>>>(mean1, 2 * NP * D / 4);
  zero_f32<<<cdiv((NP + NC) / 4, T), T, 0, stream>>>(deg_pp, (NP + NC) / 4);
  degree_count<<<cdiv(E, T), T, 0, stream>>>(pp_dst, deg_pp, E);
  degree_count<<<cdiv(E, T), T, 0, stream>>>(pc_dst, deg_pc, E);
  scatter_add4<<<cdiv(E * (D / 4), T), T, 0, stream>>>(x_product, pp_src, pp_dst, mean1, E, 6, D);
  scatter_add4<<<cdiv(E * (D / 4), T), T, 0, stream>>>(x_product, pc_src, pc_dst, mean2, E, 6, D);
  row_scale<<<cdiv(NP * (D / 4), T), T, 0, stream>>>(mean1, deg_pp, NP, 6, D);
  row_scale<<<cdiv(NC * (D / 4), T), T, 0, stream>>>(mean2, deg_pc, NC, 6, D);

  // ---- conv1: product_x = relu(mean1@Wl1^T + x_product@Wr1^T + bl1)  [NP,H]
  {
    int strips = (NP / 16) * (H / 64);
    sage_gemm<<<cdiv(strips, 8), 256, 0, stream>>>(mean1, Wl1, x_product, Wr1, bl1,
                                                   prod, NP, H, D, 1);
  }
  // ---- conv2: customer_x = relu(mean2@Wl2^T + x_customer@Wr2^T + bl2) [NC,H]
  {
    int strips = (NC / 16) * (H / 64);
    sage_gemm<<<cdiv(strips, 8), 256, 0, stream>>>(mean2, Wl2, x_customer, Wr2, bl2,
                                                   cust, NC, H, D, 1);
  }

  // ---- mean3 = scatter-mean(product_x over edge_pc)  [NC,H], reuse mean1/2 region
  float* mean3 = mean1;  // 2*NP*D == NC*H
  zero_f32<<<cdiv(NC * H / 4, T), T, 0, stream>>>(mean3, NC * H / 4);
  scatter_add4<<<cdiv(E * (H / 4), T), T, 0, stream>>>(prod, pc_src, pc_dst, mean3, E, 7, H);
  row_scale<<<cdiv(NC * (H / 4), T), T, 0, stream>>>(mean3, deg_pc, NC, 7, H);

  // ---- conv3: cust2 = relu(mean3@Wl3^T + cust@Wr3^T + bl3) [NC,H] -> reuse prod region
  float* cust2 = prod;
  {
    int strips = (NC / 16) * (H / 64);
    sage_gemm<<<cdiv(strips, 8), 256, 0, stream>>>(mean3, Wl3, cust, Wr3, bl3,
                                                   cust2, NC, H, H, 1);
  }

  // ---- final: out = cust2 @ Wlin^T + blin  [NC,O], no relu
  {
    int strips = (NC / 16) * (O / 64);
    sage_gemm<<<cdiv(strips, 8), 256, 0, stream>>>(cust2, Wlin, nullptr, nullptr, blin,
                                                   (float*)d_out, NC, O, H, 0);
  }
}